// MambaMixer_81295140978758
// MI455X (gfx1250) — compile-verified
//
#include <hip/hip_runtime.h>
#include <hip/hip_bf16.h>
#include <math.h>

// MambaMixer forward for MI455X (gfx1250), fp32 end-to-end.
// GEMMs: V_WMMA_F32_16X16X4_F32 with double-buffered LDS operand staging via
// GLOBAL_LOAD_ASYNC_TO_LDS_B128 (ASYNCcnt-tracked, overlapped with compute).
// Selective scan: 3-phase chunked associative scan (65536-way parallel).
// Workspace requirement: ~285 MB (see kernel_launch offsets).

#define B_      2
#define L_      4096
#define DMODEL  1024
#define DINNER  2048
#define DSTATE  16
#define DCONV   4
#define DTRANK  64
#define XDBLW   (DTRANK + 2 * DSTATE)   // 96
#define NCHUNK  16
#define LCHUNK  (L_ / NCHUNK)           // 256

typedef float f2  __attribute__((ext_vector_type(2)));
typedef float f4  __attribute__((ext_vector_type(4)));
typedef float v8f __attribute__((ext_vector_type(8)));

__device__ __forceinline__ float softplus_f(float v) {
  return v > 20.f ? v : log1pf(__expf(v));
}
__device__ __forceinline__ float silu_f(float v) {
  return v / (1.f + __expf(-v));
}

// 16B async global->LDS copy (GV mode: 64-bit vaddr, LDS byte addr in VDST VGPR).
// Generic shared pointers carry the LDS offset in their low 32 bits (aperture rule).
__device__ __forceinline__ void async_cp16(float* lds_dst, const float* gsrc) {
  unsigned lds_off = (unsigned)(unsigned long long)lds_dst;
  unsigned long long ga = (unsigned long long)gsrc;
  asm volatile("global_load_async_to_lds_b128 %0, %1, off"
               :: "v"(lds_off), "v"(ga) : "memory");
}
__device__ __forceinline__ void wait_async0() {
#if __has_builtin(__builtin_amdgcn_s_wait_asynccnt)
  __builtin_amdgcn_s_wait_asynccnt(0);
#else
  asm volatile("s_wait_asynccnt 0x0" ::: "memory");
#endif
}

// ---------------------------------------------------------------------------
// GEMM: C[M,N] = A[M,K] (row-major, lda) x W[N,K]^T (row-major, ldw)
// Block = 256 threads (8 waves) -> 128x128 C tile.
// K staged 16-deep in LDS, double-buffered, filled by async-to-LDS copies.
// Wave computes 64x32 via 4x2 accumulators of v_wmma_f32_16x16x4_f32.
// epi==1: C = softplus(softplus(C + bias[n]) + bias[n])   (dt path)
// ---------------------------------------------------------------------------
#define TM 128
#define TN 128
#define TK 16
#define LPAD 20   // floats per LDS row: 16B-aligned fills + conflict-free b64 frag reads

__global__ __launch_bounds__(256) void gemm_f32_wmma(
    const float* __restrict__ A, int lda,
    const float* __restrict__ W, int ldw,
    float* __restrict__ C, int ldc,
    int M, int N, int K, int epi, const float* __restrict__ bias)
{
  __shared__ float lA[2][TM * LPAD];
  __shared__ float lB[2][TN * LPAD];

  const int tid  = threadIdx.x;
  const int lane = tid & 31;
  const int wv   = tid >> 5;
  const int wRow = wv >> 2;        // 0..1  -> 64-row half of block tile
  const int wCol = wv & 3;         // 0..3  -> 32-col quarter
  const int r16  = lane & 15;      // M/N index within a 16-wide fragment
  const int kh   = lane >> 4;      // K-half select (ISA A/B 16x4 layout)
  const int m0   = blockIdx.x * TM;
  const int n0   = blockIdx.y * TN;

  // staging indices: each thread owns one 16B chunk in rows lr and lr+64
  const int lr = tid >> 2;         // 0..63
  const int lc = (tid & 3) * 4;    // 0,4,8,12
  const bool wOk0 = (n0 + lr)      < N;
  const bool wOk1 = (n0 + lr + 64) < N;

  // Ragged-N: out-of-range W rows are K-invariant -> pre-zero once, both buffers.
  if (!wOk0) {
    *(f4*)&lB[0][lr * LPAD + lc] = (f4){0.f,0.f,0.f,0.f};
    *(f4*)&lB[1][lr * LPAD + lc] = (f4){0.f,0.f,0.f,0.f};
  }
  if (!wOk1) {
    *(f4*)&lB[0][(lr + 64) * LPAD + lc] = (f4){0.f,0.f,0.f,0.f};
    *(f4*)&lB[1][(lr + 64) * LPAD + lc] = (f4){0.f,0.f,0.f,0.f};
  }

  v8f acc[4][2];
#pragma unroll
  for (int i = 0; i < 4; ++i)
#pragma unroll
    for (int j = 0; j < 2; ++j)
      acc[i][j] = (v8f){0.f,0.f,0.f,0.f,0.f,0.f,0.f,0.f};

  // prologue: fill buffer 0 with slab k0=0
  async_cp16(&lA[0][lr * LPAD + lc],        A + (size_t)(m0 + lr)      * lda + lc);
  async_cp16(&lA[0][(lr + 64) * LPAD + lc], A + (size_t)(m0 + lr + 64) * lda + lc);
  if (wOk0) async_cp16(&lB[0][lr * LPAD + lc],        W + (size_t)(n0 + lr)      * ldw + lc);
  if (wOk1) async_cp16(&lB[0][(lr + 64) * LPAD + lc], W + (size_t)(n0 + lr + 64) * ldw + lc);
  wait_async0();
  __syncthreads();

  int cur = 0;
  for (int k0 = 0; k0 < K; k0 += TK) {
    // issue next slab's async copies into the other buffer (overlaps compute)
    const int nk = k0 + TK;
    if (nk < K) {
      const int nxt = cur ^ 1;
      async_cp16(&lA[nxt][lr * LPAD + lc],        A + (size_t)(m0 + lr)      * lda + nk + lc);
      async_cp16(&lA[nxt][(lr + 64) * LPAD + lc], A + (size_t)(m0 + lr + 64) * lda + nk + lc);
      if (wOk0) async_cp16(&lB[nxt][lr * LPAD + lc],        W + (size_t)(n0 + lr)      * ldw + nk + lc);
      if (wOk1) async_cp16(&lB[nxt][(lr + 64) * LPAD + lc], W + (size_t)(n0 + lr + 64) * ldw + nk + lc);
    }

    // --- 4 x (K=4) WMMA passes over the current slab ---
#pragma unroll
    for (int kk = 0; kk < 4; ++kk) {
      const int kb = kk * 4 + kh * 2;   // lanes 0-15: K0/K1, lanes 16-31: K2/K3
      f2 afr[4];
      f2 bfr[2];
#pragma unroll
      for (int mi = 0; mi < 4; ++mi)
        afr[mi] = *(const f2*)&lA[cur][(wRow * 64 + mi * 16 + r16) * LPAD + kb];
#pragma unroll
      for (int ni = 0; ni < 2; ++ni)
        bfr[ni] = *(const f2*)&lB[cur][(wCol * 32 + ni * 16 + r16) * LPAD + kb];
#pragma unroll
      for (int mi = 0; mi < 4; ++mi)
#pragma unroll
        for (int ni = 0; ni < 2; ++ni)
          acc[mi][ni] = __builtin_amdgcn_wmma_f32_16x16x4_f32(
              false, afr[mi], false, bfr[ni], (short)0, acc[mi][ni], false, false);
    }

    wait_async0();     // next buffer's copies have landed
    __syncthreads();   // all waves done reading cur; safe to overwrite next round
    cur ^= 1;
  }

  // --- epilogue + store (C/D layout: VGPR v holds rows v and v+8) ---
#pragma unroll
  for (int mi = 0; mi < 4; ++mi) {
#pragma unroll
    for (int ni = 0; ni < 2; ++ni) {
      const int col = n0 + wCol * 32 + ni * 16 + r16;
      if (col >= N) continue;
      const float bse = (epi == 1) ? bias[col] : 0.f;
#pragma unroll
      for (int v = 0; v < 8; ++v) {
        const int row = m0 + wRow * 64 + mi * 16 + kh * 8 + v;
        float val = acc[mi][ni][v];
        if (epi == 1) val = softplus_f(softplus_f(val + bse) + bse);
        C[(size_t)row * ldc + col] = val;
      }
    }
  }
}

// ---------------------------------------------------------------------------
// Depthwise causal conv(k=4) + bias + SiLU.  proj layout: (B*L, 2*DINNER),
// u = cols [0,DINNER).  Output x: (B*L, DINNER).  Coalesced over d.
// ---------------------------------------------------------------------------
__global__ __launch_bounds__(256) void conv_silu_k(
    const float* __restrict__ proj, const float* __restrict__ cw,
    const float* __restrict__ cb, float* __restrict__ x)
{
  const int d  = blockIdx.y * 256 + threadIdx.x;
  const int b  = blockIdx.x / NCHUNK;
  const int c  = blockIdx.x % NCHUNK;
  const int l0 = c * LCHUNK;
  const float w0 = cw[d * 4 + 0], w1 = cw[d * 4 + 1];
  const float w2 = cw[d * 4 + 2], w3 = cw[d * 4 + 3];
  const float bias = cb[d];
  float u0 = (l0 >= 3) ? proj[((size_t)b * L_ + l0 - 3) * (2 * DINNER) + d] : 0.f;
  float u1 = (l0 >= 2) ? proj[((size_t)b * L_ + l0 - 2) * (2 * DINNER) + d] : 0.f;
  float u2 = (l0 >= 1) ? proj[((size_t)b * L_ + l0 - 1) * (2 * DINNER) + d] : 0.f;
  for (int l = l0; l < l0 + LCHUNK; ++l) {
    const float u3 = proj[((size_t)b * L_ + l) * (2 * DINNER) + d];
    const float v  = u0 * w0 + u1 * w1 + u2 * w2 + u3 * w3 + bias;
    x[((size_t)b * L_ + l) * DINNER + d] = silu_f(v);
    u0 = u1; u1 = u2; u2 = u3;
  }
}

// ---------------------------------------------------------------------------
// Chunked selective scan.  state_n(l) = e^{dt*A_n} state_n(l-1) + dt*u*B_n(l)
// Pass1: per (b,d,chunk) -> chunk-local (prod a, state) with zero init.
// Pass2: per (b,d,n)     -> 16-step inter-chunk scan; writes carries + last_state.
// Pass3: per (b,d,chunk) -> replay with carry, emit y, fuse +x*D and *silu(z),
//                           overwrite x in place (only this thread touches x[b,:,d] in chunk).
// ---------------------------------------------------------------------------
__global__ __launch_bounds__(256) void scan_pass1(
    const float* __restrict__ dlt, const float* __restrict__ x,
    const float* __restrict__ xdbl, const float* __restrict__ A_log,
    float* __restrict__ sumA, float* __restrict__ sumB)
{
  const int d = blockIdx.y * 256 + threadIdx.x;
  const int b = blockIdx.x / NCHUNK;
  const int c = blockIdx.x % NCHUNK;
  float Aa[DSTATE];
#pragma unroll
  for (int n = 0; n < DSTATE; ++n) Aa[n] = -__expf(A_log[d * DSTATE + n]);
  float st[DSTATE], ap[DSTATE];
#pragma unroll
  for (int n = 0; n < DSTATE; ++n) { st[n] = 0.f; ap[n] = 1.f; }
  for (int l = c * LCHUNK; l < (c + 1) * LCHUNK; ++l) {
    const size_t row = (size_t)b * L_ + l;
    const float dv = dlt[row * DINNER + d];
    const float xv = x[row * DINNER + d];
    const float du = dv * xv;
    const float* Bm = xdbl + row * XDBLW + DTRANK;
#pragma unroll
    for (int n = 0; n < DSTATE; ++n) {
      const float e = __expf(dv * Aa[n]);
      st[n] = e * st[n] + du * Bm[n];
      ap[n] *= e;
    }
  }
  const size_t o = ((size_t)(b * NCHUNK + c) * DINNER + d) * DSTATE;
#pragma unroll
  for (int n = 0; n < DSTATE; ++n) { sumA[o + n] = ap[n]; sumB[o + n] = st[n]; }
}

__global__ __launch_bounds__(256) void scan_pass2(
    const float* __restrict__ sumA, const float* __restrict__ sumB,
    float* __restrict__ carryIn, float* __restrict__ last_state)
{
  const int gid = blockIdx.x * 256 + threadIdx.x;       // 0 .. B*DINNER*DSTATE-1
  const int n = gid & (DSTATE - 1);
  const int d = (gid >> 4) & (DINNER - 1);
  const int b = gid >> 15;
  float carry = 0.f;
#pragma unroll
  for (int c = 0; c < NCHUNK; ++c) {
    const size_t o = ((size_t)(b * NCHUNK + c) * DINNER + d) * DSTATE + n;
    carryIn[o] = carry;
    carry = sumA[o] * carry + sumB[o];
  }
  last_state[((size_t)b * DINNER + d) * DSTATE + n] = carry;
}

__global__ __launch_bounds__(256) void scan_pass3(
    const float* __restrict__ dlt, float* __restrict__ x,
    const float* __restrict__ xdbl, const float* __restrict__ A_log,
    const float* __restrict__ carryIn, const float* __restrict__ proj,
    const float* __restrict__ Dp)
{
  const int d = blockIdx.y * 256 + threadIdx.x;
  const int b = blockIdx.x / NCHUNK;
  const int c = blockIdx.x % NCHUNK;
  float Aa[DSTATE];
#pragma unroll
  for (int n = 0; n < DSTATE; ++n) Aa[n] = -__expf(A_log[d * DSTATE + n]);
  float st[DSTATE];
  const size_t o = ((size_t)(b * NCHUNK + c) * DINNER + d) * DSTATE;
#pragma unroll
  for (int n = 0; n < DSTATE; ++n) st[n] = carryIn[o + n];
  const float Dd = Dp[d];
  for (int l = c * LCHUNK; l < (c + 1) * LCHUNK; ++l) {
    const size_t row = (size_t)b * L_ + l;
    const float dv = dlt[row * DINNER + d];
    const float xv = x[row * DINNER + d];
    const float du = dv * xv;
    const float* Bm = xdbl + row * XDBLW + DTRANK;
    const float* Cm = xdbl + row * XDBLW + DTRANK + DSTATE;
    float y = 0.f;
#pragma unroll
    for (int n = 0; n < DSTATE; ++n) {
      const float e = __expf(dv * Aa[n]);
      st[n] = e * st[n] + du * Bm[n];
      y += st[n] * Cm[n];
    }
    const float zv = proj[row * (2 * DINNER) + DINNER + d];
    x[row * DINNER + d] = (y + xv * Dd) * silu_f(zv);   // in-place: becomes gated y
  }
}

// conv_state[b][d][k] = u[b][L-4+k][d]
__global__ __launch_bounds__(256) void conv_state_k(
    const float* __restrict__ proj, float* __restrict__ cs)
{
  const int gid = blockIdx.x * 256 + threadIdx.x;       // (b*2048+d)*4+k
  const int k = gid & 3;
  const int d = (gid >> 2) & (DINNER - 1);
  const int b = gid >> 13;
  cs[gid] = proj[((size_t)b * L_ + (L_ - DCONV + k)) * (2 * DINNER) + d];
}

// ---------------------------------------------------------------------------
extern "C" void kernel_launch(void* const* d_in, const int* in_sizes, int n_in,
                              void* d_out, int out_size, void* d_ws, size_t ws_size,
                              hipStream_t stream)
{
  (void)in_sizes; (void)n_in; (void)out_size; (void)ws_size;
  const float* hidden     = (const float*)d_in[0];
  const float* in_proj_w  = (const float*)d_in[1];
  const float* conv_w     = (const float*)d_in[2];
  const float* conv_b     = (const float*)d_in[3];
  const float* x_proj_w   = (const float*)d_in[4];
  const float* dt_proj_w  = (const float*)d_in[5];
  const float* dt_proj_b  = (const float*)d_in[6];
  const float* A_log      = (const float*)d_in[7];
  const float* Dp         = (const float*)d_in[8];
  const float* out_proj_w = (const float*)d_in[9];

  float* out        = (float*)d_out;                               // (B,L,DMODEL)
  float* conv_state = out + (size_t)B_ * L_ * DMODEL;              // (B,DINNER,4)
  float* last_state = conv_state + (size_t)B_ * DINNER * DCONV;    // (B,DINNER,16)

  // workspace layout (floats); total ~= 71.0M floats (~285 MB)
  float* ws   = (float*)d_ws;
  float* proj = ws;                                                // (B*L, 2*DINNER)
  float* xbuf = proj + (size_t)B_ * L_ * 2 * DINNER;               // (B*L, DINNER): x, then gated y (in place)
  float* xdbl = xbuf + (size_t)B_ * L_ * DINNER;                   // (B*L, 96)
  float* dlt  = xdbl + (size_t)B_ * L_ * XDBLW;                    // (B*L, DINNER)
  float* sumA = dlt  + (size_t)B_ * L_ * DINNER;                   // (B*NC, DINNER, 16)
  float* sumB = sumA + (size_t)B_ * NCHUNK * DINNER * DSTATE;
  float* cin  = sumB + (size_t)B_ * NCHUNK * DINNER * DSTATE;

  const int M = B_ * L_;

  // 1) in_proj: proj = hidden x in_proj_w^T   (8192x1024 x 1024x4096)
  gemm_f32_wmma<<<dim3(M / TM, (2 * DINNER) / TN), 256, 0, stream>>>(
      hidden, DMODEL, in_proj_w, DMODEL, proj, 2 * DINNER,
      M, 2 * DINNER, DMODEL, 0, nullptr);

  // 2) depthwise conv + SiLU -> xbuf
  conv_silu_k<<<dim3(B_ * NCHUNK, DINNER / 256), 256, 0, stream>>>(
      proj, conv_w, conv_b, xbuf);

  // 3) x_proj: xdbl = x x x_proj_w^T  (N=96, guarded)
  gemm_f32_wmma<<<dim3(M / TM, 1), 256, 0, stream>>>(
      xbuf, DINNER, x_proj_w, DINNER, xdbl, XDBLW,
      M, XDBLW, DINNER, 0, nullptr);

  // 4) dt: delta = softplus(softplus(ts x dt_proj_w^T + b) + b)
  gemm_f32_wmma<<<dim3(M / TM, DINNER / TN), 256, 0, stream>>>(
      xdbl, XDBLW, dt_proj_w, DTRANK, dlt, DINNER,
      M, DINNER, DTRANK, 1, dt_proj_b);

  // 5) chunked selective scan (+ fused D-skip and silu(z) gating in pass3)
  scan_pass1<<<dim3(B_ * NCHUNK, DINNER / 256), 256, 0, stream>>>(
      dlt, xbuf, xdbl, A_log, sumA, sumB);
  scan_pass2<<<dim3((B_ * DINNER * DSTATE) / 256), 256, 0, stream>>>(
      sumA, sumB, cin, last_state);
  scan_pass3<<<dim3(B_ * NCHUNK, DINNER / 256), 256, 0, stream>>>(
      dlt, xbuf, xdbl, A_log, cin, proj, Dp);

  // 6) out_proj: out = y x out_proj_w^T  (8192x2048 x 2048x1024)
  gemm_f32_wmma<<<dim3(M / TM, DMODEL / TN), 256, 0, stream>>>(
      xbuf, DINNER, out_proj_w, DINNER, out, DMODEL,
      M, DMODEL, DINNER, 0, nullptr);

  // 7) conv_state tail copy
  conv_state_k<<<dim3((B_ * DINNER * DCONV) / 256), 256, 0, stream>>>(
      proj, conv_state);
}